// MetricLoss_14276471291925
// MI455X (gfx1250) — compile-verified
//
#include <hip/hip_runtime.h>
#include <hip/hip_bf16.h>

#define PP 16384   // number of points (P)
#define NN 16      // N
#define KK 4       // K neighbors

typedef __attribute__((ext_vector_type(2))) float v2f;
typedef __attribute__((ext_vector_type(8))) float v8f;

// ---------------------------------------------------------------------------
// Branchless sorted insert of (cd, cj) into ascending 4-list (values + ids)
// ---------------------------------------------------------------------------
__device__ __forceinline__ void ins4(float (&d)[4], int (&ji)[4], float cd, int cj) {
#pragma unroll
  for (int k = 0; k < 4; ++k) {
    bool lt = cd < d[k];
    float nd = lt ? cd : d[k];
    int   nj = lt ? cj : ji[k];
    float od = lt ? d[k] : cd;   // carry the max forward
    int   oj = lt ? ji[k] : cj;
    d[k] = nd; ji[k] = nj; cd = od; cj = oj;
  }
}

// ---------------------------------------------------------------------------
// Kernel 0: precompute squared norms of the P points of image 0
// ---------------------------------------------------------------------------
__global__ __launch_bounds__(256) void ml_init_kernel(const float* __restrict__ pts,
                                                      float* __restrict__ sq) {
  int i = blockIdx.x * 256 + threadIdx.x;
  if (i < PP) {
    float x = pts[3 * i + 0], y = pts[3 * i + 1], z = pts[3 * i + 2];
    sq[i] = x * x + y * y + z * z;
  }
}

// ---------------------------------------------------------------------------
// Kernel 1: KNN (4 nearest, excluding self) via V_WMMA_F32_16X16X4_F32.
// One workgroup (4 waves) per 16-row strip; each wave sweeps 256 column tiles.
// A/B tile layout (32-bit, 16x4 MxK): lanes 0-15 -> (x,y) K=0,1 of point M=lane;
// lanes 16-31 -> (z,0) K=2,3 of point M=lane-16.  C/D: lane = column, 8 rows/lane.
// Note: loads use branchless addressing; reading one float past image 0 stays
// inside the full (16,16384,3) pt_coords allocation.
// ---------------------------------------------------------------------------
__global__ __launch_bounds__(128) void ml_knn_kernel(const float* __restrict__ pts,
                                                     const float* __restrict__ sq,
                                                     int* __restrict__ nbr) {
  __shared__ float md[4096];
  __shared__ int   ms[4096];

  const int tid   = threadIdx.x;
  const int wave  = tid >> 5;
  const int lane  = tid & 31;
  const int nl    = lane & 15;
  const int sel   = lane >> 4;         // 0: (x,y) half, 1: (z,0) half
  const int rbase = sel << 3;
  const int i0    = blockIdx.x * 16;

  // A tile (rows i0..i0+15), constant for the whole strip (branchless)
  v2f a;
  {
    int base = (i0 + nl) * 3 + (sel << 1);
    float a0 = pts[base];
    float a1 = pts[base + 1];
    a[0] = a0;
    a[1] = sel ? 0.0f : a1;
  }
  // squared norms for the 8 rows this lane's C registers cover
  float sqr[8];
#pragma unroll
  for (int m = 0; m < 8; ++m) sqr[m] = sq[i0 + rbase + m];

  // running top-4 (ascending d2) per row held by this lane
  float td[8][4];
  int   ti[8][4];
#pragma unroll
  for (int m = 0; m < 8; ++m)
#pragma unroll
    for (int k = 0; k < 4; ++k) { td[m][k] = 3.0e38f; ti[m][k] = 0; }

  for (int t = wave; t < PP / 16; t += 4) {
    const int jc = (t << 4) + nl;       // this lane's column
    const float sqj = sq[jc];

    v2f b;                              // branchless tile load
    {
      int base = jc * 3 + (sel << 1);
      float b0 = pts[base];
      float b1 = pts[base + 1];
      b[0] = b0;
      b[1] = sel ? 0.0f : b1;
    }

    v8f c = {};
    c = __builtin_amdgcn_wmma_f32_16x16x4_f32(false, a, false, b,
                                              (short)0, c, false, false);

    float cand[8];
    float gate = 0.0f;                  // < 0 iff any candidate beats its t3
#pragma unroll
    for (int m = 0; m < 8; ++m) {
      float d2 = __builtin_fmaf(-2.0f, c[m], sqr[m] + sqj);
      d2 = (jc == (i0 + rbase + m)) ? 3.0e38f : d2;   // exclude self
      cand[m] = d2;
      gate = fminf(gate, d2 - td[m][3]);
    }
    if (__any(gate < 0.0f)) {
#pragma unroll
      for (int m = 0; m < 8; ++m) ins4(td[m], ti[m], cand[m], jc);
    }
  }

  // merge: each row has 4 waves x 16 lanes x 4 entries = 256 candidates
#pragma unroll
  for (int m = 0; m < 8; ++m) {
    int row = rbase + m;
    int base = row * 256 + wave * 64 + nl * 4;
#pragma unroll
    for (int k = 0; k < 4; ++k) { md[base + k] = td[m][k]; ms[base + k] = ti[m][k]; }
  }
  __syncthreads();

  if (tid < 16) {
    float bd[4] = {3.0e38f, 3.0e38f, 3.0e38f, 3.0e38f};
    int   bi[4] = {0, 0, 0, 0};
    int base = tid * 256;
    for (int q = 0; q < 256; ++q) ins4(bd, bi, md[base + q], ms[base + q]);
#pragma unroll
    for (int k = 0; k < 4; ++k) nbr[(i0 + tid) * 4 + k] = bi[k];
  }
}

// ---------------------------------------------------------------------------
// Kernel 2: per-point metric loss terms, block-reduced to per-block partials.
// partial layout: partial[c * 1024 + blockIdx.x], c in
//   {0:w, 1:nll*w, 2:klA*w, 3:klP*w, 4:klN*w}
// ---------------------------------------------------------------------------
__global__ __launch_bounds__(256) void ml_loss_kernel(const float* __restrict__ mask_pred,
                                                      const float* __restrict__ sigma,
                                                      const int* __restrict__ mask_target,
                                                      const int* __restrict__ nbr,
                                                      const int* __restrict__ k_pos,
                                                      const int* __restrict__ k_neg,
                                                      float* __restrict__ partial) {
  __shared__ float red[256];
  const int idx = blockIdx.x * 256 + threadIdx.x;   // grid sized exactly N*P
  const int n = idx >> 14;                          // / 16384
  const int p = idx & (PP - 1);

  const float* mp = mask_pred   + (size_t)n * PP;
  const int*   mt = mask_target + (size_t)n * PP;

  const int tgt = mt[p];
  const int j0_ = nbr[p * 4 + 0], j1_ = nbr[p * 4 + 1];
  const int j2_ = nbr[p * 4 + 2], j3_ = nbr[p * 4 + 3];
  const bool p0 = (mt[j0_] == tgt), p1 = (mt[j1_] == tgt);
  const bool p2 = (mt[j2_] == tgt), p3 = (mt[j3_] == tgt);
  const int cnt = (int)p0 + (int)p1 + (int)p2 + (int)p3;
  const bool point_mask = (cnt > 0) && (cnt < KK);

  const int kp = k_pos[idx], kn = k_neg[idx];
  const int  jp = (kp == 0) ? j0_ : ((kp == 1) ? j1_ : ((kp == 2) ? j2_ : j3_));
  const bool pr = (kp == 0) ? p0  : ((kp == 1) ? p1  : ((kp == 2) ? p2  : p3));
  const int  jn = (kn == 0) ? j0_ : ((kn == 1) ? j1_ : ((kn == 2) ? j2_ : j3_));
  const bool nr = !((kn == 0) ? p0 : ((kn == 1) ? p1 : ((kn == 2) ? p2 : p3)));

  const float w = (point_mask && pr && nr) ? 1.0f : 0.0f;

  const float muA = mp[p];
  const float muP = mp[jp];
  const float muN = mp[jn];
  const float v   = sigma[n];

  const float mu_t = muP * muP - muN * muN - 2.0f * muA * (muP - muN);
  const float T1 = v * v + 2.0f * muP * muP * v + 2.0f * (v + muA * muA) * (v + muP * muP)
                 - 2.0f * muA * muA * muP * muP - 4.0f * muA * muP * v;
  const float T2 = v * v + 2.0f * muN * muN * v + 2.0f * (v + muA * muA) * (v + muN * muN)
                 - 2.0f * muA * muA * muN * muN - 4.0f * muA * muN * v;
  const float T3 = 4.0f * muP * muN * v;
  const float sigma2 = 2.0f * T1 + 2.0f * T2 - 2.0f * T3;
  const float s = sqrtf(sigma2 + 1e-7f);
  const float probs = 0.5f * erfcf(mu_t / ((s + 1e-8f) * 1.41421356237309515f));
  const float nll = -logf(probs + 1e-8f);

  const float VP  = 1.0f / 96.0f;
  const float lvp = logf(VP);
  const float lv  = logf(v);
  const float base_kl = v / VP - 1.0f + lvp - lv;
  const float klA = 0.5f * (base_kl + muA * muA / VP);
  const float klP = 0.5f * (base_kl + muP * muP / VP);
  const float klN = 0.5f * (base_kl + muN * muN / VP);

  float vals[5] = {w, nll * w, klA * w, klP * w, klN * w};
#pragma unroll
  for (int c = 0; c < 5; ++c) {
    red[threadIdx.x] = vals[c];
    __syncthreads();
    for (int off = 128; off > 0; off >>= 1) {
      if (threadIdx.x < off) red[threadIdx.x] += red[threadIdx.x + off];
      __syncthreads();
    }
    if (threadIdx.x == 0) partial[c * 1024 + blockIdx.x] = red[0];
    __syncthreads();
  }
}

// ---------------------------------------------------------------------------
// Kernel 3: deterministic final reduction of 1024 partials x 5 components
// ---------------------------------------------------------------------------
__global__ __launch_bounds__(256) void ml_final_kernel(const float* __restrict__ partial,
                                                       float* __restrict__ out) {
  __shared__ float red[256];
  __shared__ float tot[5];
#pragma unroll
  for (int c = 0; c < 5; ++c) {
    float s = 0.0f;
    for (int b = threadIdx.x; b < 1024; b += 256) s += partial[c * 1024 + b];
    red[threadIdx.x] = s;
    __syncthreads();
    for (int off = 128; off > 0; off >>= 1) {
      if (threadIdx.x < off) red[threadIdx.x] += red[threadIdx.x + off];
      __syncthreads();
    }
    if (threadIdx.x == 0) tot[c] = red[0];
    __syncthreads();
  }
  if (threadIdx.x == 0) {
    float cntw = fmaxf(tot[0], 1.0f);
    out[0] = tot[1] / cntw + 1e-6f * ((tot[2] + tot[3] + tot[4]) / cntw);
  }
}

// ---------------------------------------------------------------------------
extern "C" void kernel_launch(void* const* d_in, const int* in_sizes, int n_in,
                              void* d_out, int out_size, void* d_ws, size_t ws_size,
                              hipStream_t stream) {
  const float* mask_pred   = (const float*)d_in[0];  // (16, 16384) f32
  const float* sigma       = (const float*)d_in[1];  // (16, 1)     f32
  const int*   mask_target = (const int*)  d_in[2];  // (16, 16384) i32
  const float* pts         = (const float*)d_in[3];  // (16,16384,3) f32; only [0] used
  const int*   k_pos       = (const int*)  d_in[4];  // (16, 16384) i32
  const int*   k_neg       = (const int*)  d_in[5];  // (16, 16384) i32
  float* out = (float*)d_out;

  char* ws = (char*)d_ws;
  int*   nbr     = (int*)ws;                                    // P*4 ints (256 KB)
  float* sq      = (float*)(ws + (size_t)PP * 4 * 4);           // P floats  (64 KB)
  float* partial = (float*)(ws + (size_t)PP * 4 * 4 + PP * 4);  // 5*1024 floats (20 KB)

  ml_init_kernel<<<PP / 256, 256, 0, stream>>>(pts, sq);
  ml_knn_kernel<<<PP / 16, 128, 0, stream>>>(pts, sq, nbr);
  ml_loss_kernel<<<(NN * PP) / 256, 256, 0, stream>>>(mask_pred, sigma, mask_target,
                                                      nbr, k_pos, k_neg, partial);
  ml_final_kernel<<<1, 256, 0, stream>>>(partial, out);
}